// NeighborStructureEmbedding_66090956751033
// MI455X (gfx1250) — compile-verified
//
#include <hip/hip_runtime.h>
#include <hip/hip_bf16.h>
#include <math.h>

#define N_PTS 1024
#define HID   64

typedef __attribute__((ext_vector_type(2))) float v2f;
typedef __attribute__((ext_vector_type(4))) float f4;
typedef __attribute__((ext_vector_type(8))) float v8f;

// ---------------------------------------------------------------------------
// Kernel A: per-point densities (3 scales) + kNN-entropy features (k=5,10,25),
// entropies L2-normalized per point. One thread per point; all points staged
// in LDS (12 KB).
// ---------------------------------------------------------------------------
__global__ __launch_bounds__(256) void nse_stats_kernel(
    const float* __restrict__ points, float* __restrict__ density,
    float* __restrict__ lgee)
{
    __shared__ float P[N_PTS * 3];
    const int tid = threadIdx.x;
    for (int v = tid; v < N_PTS * 3; v += 256) P[v] = points[v];
    __syncthreads();

    const int i = blockIdx.x * 256 + tid;
    const float px = P[i * 3 + 0], py = P[i * 3 + 1], pz = P[i * 3 + 2];

    float knn[26];
#pragma unroll
    for (int t = 0; t < 26; ++t) knn[t] = 3.0e38f;
    int c0 = 0, c1 = 0, c2 = 0;

    for (int j = 0; j < N_PTS; ++j) {
        float dx = px - P[j * 3 + 0];
        float dy = py - P[j * 3 + 1];
        float dz = pz - P[j * 3 + 2];
        float d2 = dx * dx + dy * dy + dz * dz;
        float dist = sqrtf(d2 > 0.f ? d2 : 1e-12f);   // _safe_sqrt
        c0 += (dist < 0.1f);
        c1 += (dist < 0.2f);
        c2 += (dist < 0.4f);
        if (dist < knn[25]) {                          // keep 26 smallest (incl. self)
            int p = 25;
            while (p > 0 && knn[p - 1] > dist) { knn[p] = knn[p - 1]; --p; }
            knn[p] = dist;
        }
    }

    // density = count / ((4/3) pi r^3); self is included (dist ~1e-6 < r)
    density[i * 3 + 0] = (float)c0 / 0.004188790204786391f;
    density[i * 3 + 1] = (float)c1 / 0.033510321638291124f;
    density[i * 3 + 2] = (float)c2 / 0.2680825731063290f;

    // entropies over knn[1..k] (drop self at index 0)
    const int KV[3] = {5, 10, 25};
    float e[3];
#pragma unroll
    for (int ki = 0; ki < 3; ++ki) {
        const int K = KV[ki];
        float ssum = 0.f;
        for (int t = 1; t <= K; ++t) ssum += knn[t];
        float ee = 0.f;
        for (int t = 1; t <= K; ++t) {
            float p = knn[t] / ssum;
            ee -= p * logf(p + 1e-10f);
        }
        e[ki] = ee;
    }
    float inv = 1.f / fmaxf(sqrtf(e[0] * e[0] + e[1] * e[1] + e[2] * e[2]), 1e-12f);
    lgee[i * 3 + 0] = e[0] * inv;
    lgee[i * 3 + 1] = e[1] * inv;
    lgee[i * 3 + 2] = e[2] * inv;
}

// ---------------------------------------------------------------------------
// Kernel B: per-scale moments of density. mu[s] = mean, V[s] = sum (d-mu)^2.
// Double accumulation so the variance expansion doesn't cancel.
// ---------------------------------------------------------------------------
__global__ __launch_bounds__(256) void nse_moments_kernel(
    const float* __restrict__ density, float* __restrict__ muV)
{
    __shared__ double sh[256];
    const int tid = threadIdx.x;
    for (int s = 0; s < 3; ++s) {
        double s1 = 0.0, s2 = 0.0;
        for (int j = tid; j < N_PTS; j += 256) {
            double d = (double)density[j * 3 + s];
            s1 += d; s2 += d * d;
        }
        sh[tid] = s1; __syncthreads();
        for (int o = 128; o > 0; o >>= 1) { if (tid < o) sh[tid] += sh[tid + o]; __syncthreads(); }
        double S1 = sh[0]; __syncthreads();
        sh[tid] = s2; __syncthreads();
        for (int o = 128; o > 0; o >>= 1) { if (tid < o) sh[tid] += sh[tid + o]; __syncthreads(); }
        double S2 = sh[0]; __syncthreads();
        if (tid == 0) {
            double mu = S1 / (double)N_PTS;
            double V  = S2 - S1 * S1 / (double)N_PTS;
            if (V < 0.0) V = 0.0;
            muV[s]     = (float)mu;
            muV[4 + s] = (float)V;
        }
    }
}

// ---------------------------------------------------------------------------
// Kernel C: WMMA embed.  s[m, :] = l2norm(mdve[m] @ W1 + b1) + l2norm(lgee[m] @ W2 + b2)
// Block = 128 threads (4 waves); each wave owns a 16-row strip, 4 N-tiles of 16.
// Uses V_WMMA_F32_16X16X4_F32 with K padded to 8 (mdve) / 4 (lgee).
//
// Fragment layouts (ISA 7.12.2, 32-bit 16x16x4):
//   A: lane l, vgpr v -> A[M = l%16][K = (l/16)*2 + v]
//   B: lane l, vgpr v -> B[K = (l/16)*2 + v][N = l%16]
//   C/D: lane l, vgpr r -> D[M = r + 8*(l/16)][N = l%16]
// ---------------------------------------------------------------------------
__global__ __launch_bounds__(128) void nse_embed_kernel(
    const float* __restrict__ density, const float* __restrict__ lgee,
    const float* __restrict__ muV,
    const float* __restrict__ W1, const float* __restrict__ b1,
    const float* __restrict__ W2, const float* __restrict__ b2,
    float* __restrict__ sOut)
{
    __shared__ float A1[64][8];   // l2norm([density, gradient]) padded K=8
    __shared__ float A2[64][4];   // lgee padded K=4
    const int tid = threadIdx.x;
    const int m0  = blockIdx.x * 64;

    if (tid < 64) {
        const int i = m0 + tid;
        float d[3], g[3];
        float ss = 0.f;
#pragma unroll
        for (int s = 0; s < 3; ++s) {
            d[s] = density[i * 3 + s];
            float dm = d[s] - muV[s];
            float x  = 1024.0f * dm * dm + muV[4 + s];  // == sum_j (d_i - d_j)^2
            g[s] = sqrtf(x > 0.f ? x : 1e-12f);         // _safe_sqrt
            ss += d[s] * d[s] + g[s] * g[s];
        }
        float inv = 1.f / fmaxf(sqrtf(ss), 1e-12f);
#pragma unroll
        for (int s = 0; s < 3; ++s) {
            A1[tid][s]     = d[s] * inv;
            A1[tid][3 + s] = g[s] * inv;
        }
        A1[tid][6] = 0.f; A1[tid][7] = 0.f;
        A2[tid][0] = lgee[i * 3 + 0];
        A2[tid][1] = lgee[i * 3 + 1];
        A2[tid][2] = lgee[i * 3 + 2];
        A2[tid][3] = 0.f;
    }
    __syncthreads();

    const int lane    = tid & 31;
    const int rowBase = (tid >> 5) * 16;
    const int ml      = lane & 15;           // tile row for A, tile col for B/C/D
    const int kb      = (lane >> 4) * 2;     // K base per half-wave

    // A fragments (uniform across tiles)
    v2f a1_0 = { A1[rowBase + ml][kb],     A1[rowBase + ml][kb + 1] };     // K 0..3
    v2f a1_1 = { A1[rowBase + ml][4 + kb], A1[rowBase + ml][4 + kb + 1] }; // K 4..7 (6,7 zero)
    v2f a2_0 = { A2[rowBase + ml][kb],     A2[rowBase + ml][kb + 1] };     // K 0..3 (3 zero)

    v8f acc1[4], acc2[4];
#pragma unroll
    for (int t = 0; t < 4; ++t) {
        const int n = t * 16 + ml;
        // --- mdve @ W1 ---
        v2f bA = { W1[(kb + 0) * HID + n], W1[(kb + 1) * HID + n] };       // K 0..3 all valid
        const int k4 = 4 + kb;
        v2f bB = { (k4     < 6) ? W1[(k4)     * HID + n] : 0.f,
                   (k4 + 1 < 6) ? W1[(k4 + 1) * HID + n] : 0.f };
        v8f c = {};
        c = __builtin_amdgcn_wmma_f32_16x16x4_f32(false, a1_0, false, bA, (short)0, c, false, false);
        c = __builtin_amdgcn_wmma_f32_16x16x4_f32(false, a1_1, false, bB, (short)0, c, false, false);
        const float bias1 = b1[n];
#pragma unroll
        for (int r = 0; r < 8; ++r) c[r] += bias1;
        acc1[t] = c;

        // --- lgee @ W2 ---
        v2f bC = { (kb     < 3) ? W2[(kb)     * HID + n] : 0.f,
                   (kb + 1 < 3) ? W2[(kb + 1) * HID + n] : 0.f };
        v8f c2 = {};
        c2 = __builtin_amdgcn_wmma_f32_16x16x4_f32(false, a2_0, false, bC, (short)0, c2, false, false);
        const float bias2 = b2[n];
#pragma unroll
        for (int r = 0; r < 8; ++r) c2[r] += bias2;
        acc2[t] = c2;
    }

    // Row L2-norms: reduce over the 64 N-columns.  For vgpr r, lanes 0-15 hold
    // row M=r, lanes 16-31 hold row M=r+8 -> xor-reduce within 16-lane halves.
#pragma unroll
    for (int r = 0; r < 8; ++r) {
        float s1 = 0.f, s2 = 0.f;
#pragma unroll
        for (int t = 0; t < 4; ++t) {
            s1 += acc1[t][r] * acc1[t][r];
            s2 += acc2[t][r] * acc2[t][r];
        }
#pragma unroll
        for (int m = 1; m <= 8; m <<= 1) {
            s1 += __shfl_xor(s1, m, 32);
            s2 += __shfl_xor(s2, m, 32);
        }
        const float i1 = 1.f / fmaxf(sqrtf(s1), 1e-12f);
        const float i2 = 1.f / fmaxf(sqrtf(s2), 1e-12f);
        const int mrow = m0 + rowBase + r + ((lane >> 4) << 3);
#pragma unroll
        for (int t = 0; t < 4; ++t) {
            const int n = t * 16 + ml;
            sOut[mrow * HID + n] = acc1[t][r] * i1 + acc2[t][r] * i2;
        }
    }
}

// ---------------------------------------------------------------------------
// Kernel D: out[i, j, h] = s[j, h]  -> replicate the 256 KB s-block 1024x.
// 256 MB of non-temporal stores (output > 192 MB L2); s stays L2-resident.
// ---------------------------------------------------------------------------
__global__ __launch_bounds__(256) void nse_bcast_kernel(
    const f4* __restrict__ s4, f4* __restrict__ out4)
{
    const unsigned idx = blockIdx.x * 256u + threadIdx.x;   // 16,777,216 f4 total
    f4 v = s4[idx & 16383u];                                // 1024*64/4 = 16384 f4 per slab
    __builtin_nontemporal_store(v, &out4[idx]);
}

// ---------------------------------------------------------------------------
extern "C" void kernel_launch(void* const* d_in, const int* in_sizes, int n_in,
                              void* d_out, int out_size, void* d_ws, size_t ws_size,
                              hipStream_t stream)
{
    (void)in_sizes; (void)n_in; (void)out_size; (void)ws_size;
    const float* points = (const float*)d_in[0];
    // d_in[1] = W_rtdie, d_in[2] = b_rtdie: do not affect the output.
    const float* W_mdve = (const float*)d_in[3];
    const float* b_mdve = (const float*)d_in[4];
    const float* W_lgee = (const float*)d_in[5];
    const float* b_lgee = (const float*)d_in[6];
    float* out = (float*)d_out;

    float* ws      = (float*)d_ws;
    float* density = ws;            // 1024*3
    float* lgee    = ws + 3072;     // 1024*3
    float* muV     = ws + 6144;     // mu[3] @0..2, V[3] @4..6 (pad to 16)
    float* sOut    = ws + 6160;     // 1024*64 (16B aligned: 6160*4 % 16 == 0)

    nse_stats_kernel<<<N_PTS / 256, 256, 0, stream>>>(points, density, lgee);
    nse_moments_kernel<<<1, 256, 0, stream>>>(density, muV);
    nse_embed_kernel<<<N_PTS / 64, 128, 0, stream>>>(density, lgee, muV,
                                                     W_mdve, b_mdve, W_lgee, b_lgee, sOut);
    nse_bcast_kernel<<<(N_PTS * N_PTS * HID / 4) / 256, 256, 0, stream>>>(
        (const f4*)sOut, (f4*)out);
}